// ChebyNet_88433376625309
// MI455X (gfx1250) — compile-verified
//
#include <hip/hip_runtime.h>
#include <hip/hip_bf16.h>

typedef float v2f __attribute__((ext_vector_type(2)));
typedef float v8f __attribute__((ext_vector_type(8)));

#define B_ 4
#define N_ 1024
#define K_ 100
#define T_ 2
#define H_ 64
#define F_ 96

static __device__ inline v8f zero8() {
    v8f z;
#pragma unroll
    for (int i = 0; i < 8; ++i) z[i] = 0.f;
    return z;
}

static __device__ inline v8f wmma_f32(v2f a, v2f b, v8f c) {
    // D(16x16) = A(16x4) * B(4x16) + C ; f32 everywhere
    return __builtin_amdgcn_wmma_f32_16x16x4_f32(false, a, false, b, (short)0, c,
                                                 false, false);
}

// ---------------------------------------------------------------------------
// Kernel 0: zero the whole output buffer (Etot/Force/Virial are accumulated
// with atomics; Ei is overwritten but zeroing it too is harmless).
// ---------------------------------------------------------------------------
__global__ void zero_kernel(float* __restrict__ out, int n) {
    int i = blockIdx.x * blockDim.x + threadIdx.x;
    if (i < n) out[i] = 0.f;
}

// ---------------------------------------------------------------------------
// Kernel 1: per-atom MLP forward + backward via WMMA f32 16x16x4.
// One wave handles a 16-atom tile. Computes for both types t=0,1 and masks
// results with Imagetype_map (matches reference semantics exactly).
// Outputs: Ei (B,N), atomic Etot (B), dE = dEi/dfeat (B,N,96) into workspace.
// ---------------------------------------------------------------------------
__global__ __launch_bounds__(128) void mlp_kernel(
    const float* __restrict__ feat,   // (B,N,96)
    const int*   __restrict__ typ,    // (N)
    const float* __restrict__ W1,     // (T,96,64)
    const float* __restrict__ b1,     // (T,64)
    const float* __restrict__ W2,     // (T,64,64)
    const float* __restrict__ b2,     // (T,64)
    const float* __restrict__ rdt,    // (T,64)
    const float* __restrict__ Wout,   // (T,64)
    const float* __restrict__ bout,   // (T)
    float* __restrict__ Ei_out,       // (B,N)
    float* __restrict__ Etot,         // (B)
    float* __restrict__ dE)           // (B,N,96) workspace
{
    __shared__ float lds[4][16 * 66];   // per-wave 16x64 tile, padded stride 66

    const int wave = threadIdx.x >> 5;
    const int lane = threadIdx.x & 31;
    const int tile = blockIdx.x * 4 + wave;   // 256 tiles of 16 atoms
    const int a0   = tile * 16;
    const int b    = a0 / N_;
    const int n0   = a0 % N_;
    float* sm = lds[wave];

    const int lr = lane & 15;     // row / column index within 16
    const int hi = lane >> 4;     // lane half selects K pair {0,1} vs {2,3}
    const int myType = typ[n0 + lr];   // type of row `lr` (dup on upper half)

    const float* featRow = feat + (size_t)(b * N_ + n0 + lr) * F_;

    for (int t = 0; t < T_; ++t) {
        const float* W1t   = W1   + t * F_ * H_;
        const float* W2t   = W2   + t * H_ * H_;
        const float* b1t   = b1   + t * H_;
        const float* b2t   = b2   + t * H_;
        const float* rdtt  = rdt  + t * H_;
        const float* Woutt = Wout + t * H_;

        // ---------------- Z1 = feat(16x96) @ W1t(96x64) ----------------
        v8f acc1[4];
#pragma unroll
        for (int j = 0; j < 4; ++j) acc1[j] = zero8();
        for (int kk = 0; kk < F_ / 4; ++kk) {
            const int kA = kk * 4 + hi * 2;
            v2f a = *(const v2f*)(featRow + kA);
#pragma unroll
            for (int j = 0; j < 4; ++j) {
                const int nj = j * 16 + lr;
                v2f bf;
                bf.x = W1t[kA * H_ + nj];
                bf.y = W1t[(kA + 1) * H_ + nj];
                acc1[j] = wmma_f32(a, bf, acc1[j]);
            }
        }
        // h1 = tanh(Z1 + b1) -> regs (acc1) and LDS tile
#pragma unroll
        for (int j = 0; j < 4; ++j) {
            const int nj = j * 16 + lr;
            const float bb = b1t[nj];
#pragma unroll
            for (int r = 0; r < 8; ++r) {
                const float h = tanhf(acc1[j][r] + bb);
                acc1[j][r] = h;
                sm[(r + hi * 8) * 66 + nj] = h;
            }
        }

        // ---------------- Z2 = h1(16x64) @ W2t(64x64) ----------------
        v8f acc2[4];
#pragma unroll
        for (int j = 0; j < 4; ++j) acc2[j] = zero8();
        for (int kk = 0; kk < H_ / 4; ++kk) {
            const int kA = kk * 4 + hi * 2;
            v2f a = *(const v2f*)(sm + lr * 66 + kA);
#pragma unroll
            for (int j = 0; j < 4; ++j) {
                const int nj = j * 16 + lr;
                v2f bf;
                bf.x = W2t[kA * H_ + nj];
                bf.y = W2t[(kA + 1) * H_ + nj];
                acc2[j] = wmma_f32(a, bf, acc2[j]);
            }
        }
        // u = tanh(Z2+b2); h2 = h1 + rdt*u -> LDS ; dz2 = rdt*Wout*(1-u^2) -> acc2
#pragma unroll
        for (int j = 0; j < 4; ++j) {
            const int nj = j * 16 + lr;
            const float bb = b2t[nj], rd = rdtt[nj], wo = Woutt[nj];
#pragma unroll
            for (int r = 0; r < 8; ++r) {
                const float u  = tanhf(acc2[j][r] + bb);
                const float h2 = acc1[j][r] + rd * u;
                sm[(r + hi * 8) * 66 + nj] = h2;
                acc2[j][r] = rd * wo * (1.f - u * u);
            }
        }

        // e[row] = h2[row,:] . Wout + bout  (lanes 0..15 own rows)
        float e = 0.f;
        for (int c = 0; c < H_; ++c) e += sm[lr * 66 + c] * Woutt[c];
        const float ev = e + bout[t];
        if (lane < 16 && myType == t) Ei_out[(size_t)b * N_ + n0 + lr] = ev;
        float red = (lane < 16 && myType == t) ? ev : 0.f;
#pragma unroll
        for (int off = 16; off; off >>= 1) red += __shfl_xor(red, off, 32);
        if (lane == 0) atomicAdd(Etot + b, red);

        // dz2 -> LDS (overwrite h2)
#pragma unroll
        for (int j = 0; j < 4; ++j) {
            const int nj = j * 16 + lr;
#pragma unroll
            for (int r = 0; r < 8; ++r) sm[(r + hi * 8) * 66 + nj] = acc2[j][r];
        }

        // ---------------- dh1 = dz2(16x64) @ W2t^T(64x64) ----------------
        v8f acc3[4];
#pragma unroll
        for (int j = 0; j < 4; ++j) acc3[j] = zero8();
        for (int kk = 0; kk < H_ / 4; ++kk) {
            const int kA = kk * 4 + hi * 2;
            v2f a = *(const v2f*)(sm + lr * 66 + kA);
#pragma unroll
            for (int j = 0; j < 4; ++j) {
                const int nj = j * 16 + lr;
                v2f bf = *(const v2f*)(W2t + nj * H_ + kA);  // W2[n][k] = B[k][n]
                acc3[j] = wmma_f32(a, bf, acc3[j]);
            }
        }
        // dz1 = (Wout + dh1) * (1 - h1^2) -> LDS
#pragma unroll
        for (int j = 0; j < 4; ++j) {
            const int nj = j * 16 + lr;
            const float wo = Woutt[nj];
#pragma unroll
            for (int r = 0; r < 8; ++r) {
                const float h1v = acc1[j][r];
                sm[(r + hi * 8) * 66 + nj] =
                    (wo + acc3[j][r]) * (1.f - h1v * h1v);
            }
        }

        // ---------------- dE = dz1(16x64) @ W1t^T(64x96) ----------------
        v8f accE[6];
#pragma unroll
        for (int j = 0; j < 6; ++j) accE[j] = zero8();
        for (int kk = 0; kk < H_ / 4; ++kk) {
            const int kA = kk * 4 + hi * 2;
            v2f a = *(const v2f*)(sm + lr * 66 + kA);
#pragma unroll
            for (int j = 0; j < 6; ++j) {
                const int fj = j * 16 + lr;
                v2f bf = *(const v2f*)(W1t + fj * H_ + kA);  // W1[f][k] = B[k][f]
                accE[j] = wmma_f32(a, bf, accE[j]);
            }
        }
        // masked store of dE rows whose atom type == t
#pragma unroll
        for (int j = 0; j < 6; ++j) {
            const int fj = j * 16 + lr;
#pragma unroll
            for (int r = 0; r < 8; ++r) {
                const int m  = r + hi * 8;
                const int tm = __shfl(myType, m, 32);
                if (tm == t)
                    dE[(size_t)(b * N_ + n0 + m) * F_ + fj] = accE[j][r];
            }
        }
    }
}

// ---------------------------------------------------------------------------
// Kernel 2: pair_f = dE . dfeat (masked), force scatter, center force, virial.
// One block per (b,n); 4 waves; wave w handles k = w, w+4, ...
// Streams 472 MB of dfeat with non-temporal coalesced loads -> HBM-bound.
// ---------------------------------------------------------------------------
__global__ __launch_bounds__(128) void pair_kernel(
    const float* __restrict__ dfeat,    // (B,N,K,96,3)
    const float* __restrict__ ImageDR,  // (B,N,K,4)
    const int*   __restrict__ neigh,    // (B,N,K)
    const float* __restrict__ dE,       // (B,N,96)
    float* __restrict__ Force,          // (B,N,3)  [d_out region]
    float* __restrict__ Virial)         // (B,9)    [d_out region]
{
    const int bn   = blockIdx.x;        // 0..B*N-1
    const int b    = bn / N_;
    const int n    = bn % N_;
    const int wave = threadIdx.x >> 5;
    const int lane = threadIdx.x & 31;

    const float* dErow = dE + (size_t)bn * F_;
    const float de0 = dErow[lane];
    const float de1 = dErow[lane + 32];
    const float de2 = dErow[lane + 64];

    const float* dfBase = dfeat + (size_t)bn * K_ * F_ * 3;
    const float* drBase = ImageDR + (size_t)bn * K_ * 4;
    const int*   nbBase = neigh + (size_t)bn * K_;

    float cs0 = 0.f, cs1 = 0.f, cs2 = 0.f;
    float v00 = 0.f, v01 = 0.f, v02 = 0.f;
    float v10 = 0.f, v11 = 0.f, v12 = 0.f;
    float v20 = 0.f, v21 = 0.f, v22 = 0.f;

    for (int k = wave; k < K_; k += 4) {
        const float* df = dfBase + (size_t)k * (F_ * 3);
        float p0 = 0.f, p1 = 0.f, p2 = 0.f;
        {
            const float* p = df + lane * 3;
            p0 += de0 * __builtin_nontemporal_load(p + 0);
            p1 += de0 * __builtin_nontemporal_load(p + 1);
            p2 += de0 * __builtin_nontemporal_load(p + 2);
            p = df + (lane + 32) * 3;
            p0 += de1 * __builtin_nontemporal_load(p + 0);
            p1 += de1 * __builtin_nontemporal_load(p + 1);
            p2 += de1 * __builtin_nontemporal_load(p + 2);
            p = df + (lane + 64) * 3;
            p0 += de2 * __builtin_nontemporal_load(p + 0);
            p1 += de2 * __builtin_nontemporal_load(p + 1);
            p2 += de2 * __builtin_nontemporal_load(p + 2);
        }
#pragma unroll
        for (int off = 16; off; off >>= 1) {
            p0 += __shfl_xor(p0, off, 32);
            p1 += __shfl_xor(p1, off, 32);
            p2 += __shfl_xor(p2, off, 32);
        }
        const int idx = nbBase[k];
        const float valid = (idx > 0) ? 1.f : 0.f;
        p0 *= valid; p1 *= valid; p2 *= valid;

        if (idx > 0 && lane < 3) {
            const float pv = (lane == 0) ? p0 : ((lane == 1) ? p1 : p2);
            atomicAdd(&Force[((size_t)b * N_ + (idx - 1)) * 3 + lane], pv);
        }
        cs0 += p0; cs1 += p1; cs2 += p2;

        const float rx = drBase[k * 4 + 1];
        const float ry = drBase[k * 4 + 2];
        const float rz = drBase[k * 4 + 3];
        v00 += rx * p0; v01 += rx * p1; v02 += rx * p2;
        v10 += ry * p0; v11 += ry * p1; v12 += ry * p2;
        v20 += rz * p0; v21 += rz * p1; v22 += rz * p2;
    }

    // block reduction of virial (9) + center-force sum (3)
    __shared__ float red[4][12];
    if (lane < 12) {
        float val;
        switch (lane) {
            case 0: val = v00; break;  case 1: val = v01; break;
            case 2: val = v02; break;  case 3: val = v10; break;
            case 4: val = v11; break;  case 5: val = v12; break;
            case 6: val = v20; break;  case 7: val = v21; break;
            case 8: val = v22; break;  case 9: val = cs0; break;
            case 10: val = cs1; break; default: val = cs2; break;
        }
        red[wave][lane] = val;
    }
    __syncthreads();
    if (threadIdx.x < 12) {
        const float s = red[0][threadIdx.x] + red[1][threadIdx.x] +
                        red[2][threadIdx.x] + red[3][threadIdx.x];
        if (threadIdx.x < 9)
            atomicAdd(&Virial[b * 9 + threadIdx.x], -s);
        else
            atomicAdd(&Force[((size_t)b * N_ + n) * 3 + (threadIdx.x - 9)], -s);
    }
}

// ---------------------------------------------------------------------------
extern "C" void kernel_launch(void* const* d_in, const int* in_sizes, int n_in,
                              void* d_out, int out_size, void* d_ws, size_t ws_size,
                              hipStream_t stream) {
    const float* feat    = (const float*)d_in[0];
    const float* dfeat   = (const float*)d_in[1];
    const float* ImageDR = (const float*)d_in[2];
    const int*   neigh   = (const int*)d_in[3];
    const int*   typ     = (const int*)d_in[4];
    const float* W1      = (const float*)d_in[5];
    const float* b1      = (const float*)d_in[6];
    const float* W2      = (const float*)d_in[7];
    const float* b2      = (const float*)d_in[8];
    const float* rdt     = (const float*)d_in[9];
    const float* Wout    = (const float*)d_in[10];
    const float* bout    = (const float*)d_in[11];
    // d_in[12] = nghost (0)

    float* out    = (float*)d_out;
    float* Etot   = out;                         // B
    float* Ei     = out + B_;                    // B*N
    float* Force  = out + B_ + B_ * N_;          // B*N*3
    float* Virial = out + B_ + B_ * N_ + B_ * N_ * 3;  // B*9
    float* dE     = (float*)d_ws;                // B*N*96 floats

    const int total_out = B_ + B_ * N_ + B_ * N_ * 3 + B_ * 9;
    zero_kernel<<<(total_out + 255) / 256, 256, 0, stream>>>(out, total_out);

    // 256 tiles of 16 atoms, 4 waves (tiles) per block
    mlp_kernel<<<64, 128, 0, stream>>>(feat, typ, W1, b1, W2, b2, rdt, Wout,
                                       bout, Ei, Etot, dE);

    pair_kernel<<<B_ * N_, 128, 0, stream>>>(dfeat, ImageDR, neigh, dE, Force,
                                             Virial);
}